// ViscoelasticWave2D_49306224558354
// MI455X (gfx1250) — compile-verified
//
#include <hip/hip_runtime.h>
#include <hip/hip_bf16.h>
#include <math.h>

typedef __attribute__((ext_vector_type(2))) float v2f;
typedef __attribute__((ext_vector_type(8))) float v8f;

#define NSTEPS 300
#define NBLOCKS 32

// physics constants (f32, matching reference)
#define DXc      (7.8125e-4f)      /* 0.2/256 */
#define INV_Dc   (1280.0f)
#define DTc      (2.5e-5f)
#define MUc      (25000.0f)        /* rho*cs^2 */
#define ETAc     (0.1f)
#define DT_RHOc  (2.5e-8f)         /* DT/RHO */
#define BULKFc   (0.99999975f)     /* 1 - 0.01*DT */
#define TWOSIG2c (1.8e-5f)         /* 2*sigma^2 */
#define EPSc     (1e-8f)

__device__ __forceinline__ void grid_sync(unsigned* cnt, unsigned* gen) {
    // make this thread's global stores device-visible before signalling
    asm volatile("s_wait_storecnt 0" ::: "memory");
    __syncthreads();
    if (threadIdx.x == 0) {
        __threadfence();
        unsigned g = __hip_atomic_load(gen, __ATOMIC_RELAXED, __HIP_MEMORY_SCOPE_AGENT);
        unsigned arrived = __hip_atomic_fetch_add(cnt, 1u, __ATOMIC_ACQ_REL, __HIP_MEMORY_SCOPE_AGENT);
        if (arrived == NBLOCKS - 1u) {
            __hip_atomic_store(cnt, 0u, __ATOMIC_RELAXED, __HIP_MEMORY_SCOPE_AGENT);
            __hip_atomic_store(gen, g + 1u, __ATOMIC_RELEASE, __HIP_MEMORY_SCOPE_AGENT);
        } else {
            while (__hip_atomic_load(gen, __ATOMIC_ACQUIRE, __HIP_MEMORY_SCOPE_AGENT) == g) {
                __builtin_amdgcn_s_sleep(2);
            }
        }
        __threadfence();   // acquire: invalidate WGP$ so halo loads see fresh L2 data
    }
    __syncthreads();
}

__global__ void __launch_bounds__(256, 1)
wave_init_kernel(float* ws) {
    int i = blockIdx.x * blockDim.x + threadIdx.x;
    if (i < 16400) ws[i] = 0.0f;   // halos (4*4096) + barrier counters (bit pattern 0)
}

__global__ void __launch_bounds__(256, 1)
wave_persistent_kernel(const float* __restrict__ traj,
                       float* __restrict__ out,
                       float* __restrict__ ws) {
    float* haloVzRow = ws;              // [256 tiles][16]  row 0 of each tile's vz
    float* haloVzCol = ws + 4096;       // col 0 of each tile's vz
    float* haloSxz   = ws + 8192;       // row 15 of each tile's sigma_xz
    float* haloSyz   = ws + 12288;      // col 15 of each tile's sigma_yz
    unsigned* barCnt = (unsigned*)(ws + 16384);
    unsigned* barGen = barCnt + 1;

    __shared__ float lds[8][2][256];    // per-wave 16x16 staging tiles (16 KB)

    const int tid  = threadIdx.x;
    const int wave = tid >> 5;
    const int lane = tid & 31;
    const int hi   = lane >> 4;         // half-wave select
    const int n    = lane & 15;         // N (column) index / A-matrix M index
    const int tile = blockIdx.x * 8 + wave;   // 0..255
    const int ti   = tile >> 4;         // tile row (x / i)
    const int tj   = tile & 15;         // tile col (y / j)
    const int mb   = 8 * hi;            // C-layout: VGPR v holds M = v + mb

    float* ld0 = &lds[wave][0][0];
    float* ld1 = &lds[wave][1][0];

    // ---- per-lane constant difference-matrix operand chunks (K=4 each) ----
    // A layout 16x4: lanes0-15 hold (K=c0,c0+1), lanes16-31 (K=c0,c0+1) with c0=4k+2*hi, M = n
    // B layout 4x16: lanes0-15 rows (r0,r0+1), lanes16-31 rows (r0,r0+1) with r0=4k+2*hi, N = n
    v2f aLx[4], aLb[4], bRy[4], bRb[4];
#pragma unroll
    for (int k = 0; k < 4; ++k) {
        int c0 = 4 * k + 2 * hi;
#pragma unroll
        for (int e = 0; e < 2; ++e) {
            int c = c0 + e;
            // forward diff rows: L[m,m]=-1 (0 at global last row), L[m,m+1]=+1
            float lx = (c == n) ? ((ti == 15 && n == 15) ? 0.0f : -1.0f)
                                : ((c == n + 1) ? 1.0f : 0.0f);
            // backward diff rows: L[m,m]=+1 (0 at global first row), L[m,m-1]=-1
            float lb = (c == n) ? ((ti == 0 && n == 0) ? 0.0f : 1.0f)
                                : ((c == n - 1) ? -1.0f : 0.0f);
            // forward diff cols: R[j,j]=-1 (0 at global last col), R[j+1,j]=+1
            float ry = (c == n) ? ((tj == 15 && n == 15) ? 0.0f : -1.0f)
                                : ((c == n + 1) ? 1.0f : 0.0f);
            // backward diff cols: R[j,j]=+1 (0 at global first col), R[j-1,j]=-1
            float rb = (c == n) ? ((tj == 0 && n == 0) ? 0.0f : 1.0f)
                                : ((n == c + 1) ? -1.0f : 0.0f);
            if (e == 0) { aLx[k].x = lx; aLb[k].x = lb; bRy[k].x = ry; bRb[k].x = rb; }
            else        { aLx[k].y = lx; aLb[k].y = lb; bRy[k].y = ry; bRb[k].y = rb; }
        }
    }

    // ---- per-lane PML damping & source grid coordinates ----
    auto prof = [](int i) -> float {
        int e = (i < 128) ? i : 255 - i;
        if (e < 20)
            return 1.0f - 0.05f * (0.5f + 0.5f * cosf(3.14159265358979f * (float)(20 - e) / 20.0f));
        return 1.0f;
    };
    const int   gj  = tj * 16 + n;
    const float gy  = ((float)gj + 0.5f) * DXc;
    const float py  = prof(gj);
    float damp[8], gxv[8];
#pragma unroll
    for (int v = 0; v < 8; ++v) {
        int gi = ti * 16 + mb + v;
        damp[v] = prof(gi) * py;
        gxv[v]  = ((float)gi + 0.5f) * DXc;
    }
    const int gbase = (ti * 16 + mb) * 256 + gj;   // + v*256 per VGPR row

    float* vzHist = out;
    float* stHist = out + (size_t)NSTEPS * 65536;

    v8f vz = {}, exz = {}, eyz = {};

#pragma unroll 1
    for (int s = 0; s < NSTEPS; ++s) {
        const float xt = traj[2 * s];
        const float yt = traj[2 * s + 1];

        // ================= phase A: strain / stress =================
#pragma unroll
        for (int v = 0; v < 8; ++v) ld0[(mb + v) * 16 + n] = vz[v];
        asm volatile("s_wait_dscnt 0" ::: "memory");

        v2f bvz[4], avz[4];
#pragma unroll
        for (int k = 0; k < 4; ++k) {
            int r0 = 4 * k + 2 * hi;
            bvz[k].x = ld0[r0 * 16 + n];
            bvz[k].y = ld0[(r0 + 1) * 16 + n];
            avz[k].x = ld0[n * 16 + r0];
            avz[k].y = ld0[n * 16 + r0 + 1];
        }
        v8f dvx = {}, dvy = {};
#pragma unroll
        for (int k = 0; k < 4; ++k)
            dvx = __builtin_amdgcn_wmma_f32_16x16x4_f32(false, aLx[k], false, bvz[k],
                                                        (short)0, dvx, false, false);
#pragma unroll
        for (int k = 0; k < 4; ++k)
            dvy = __builtin_amdgcn_wmma_f32_16x16x4_f32(false, avz[k], false, bRy[k],
                                                        (short)0, dvy, false, false);
        // halo fixes: row 15 gets +vz(below tile, row0); col 15 gets +vz(right tile, col0)
        if (ti < 15 && hi) dvx[7] += haloVzRow[((ti + 1) * 16 + tj) * 16 + n];
        if (tj < 15 && n == 15) {
            const float* h = &haloVzCol[(ti * 16 + tj + 1) * 16 + mb];
#pragma unroll
            for (int v = 0; v < 8; ++v) dvy[v] += h[v];
        }

        v8f sxz, syz;
#pragma unroll
        for (int v = 0; v < 8; ++v) {
            float dx = dvx[v] * INV_Dc;
            float dy = dvy[v] * INV_Dc;
            exz[v] += dx * DTc;
            eyz[v] += dy * DTc;
            sxz[v] = MUc * exz[v] + ETAc * dx;
            syz[v] = MUc * eyz[v] + ETAc * dy;
            stHist[(size_t)s * 65536 + gbase + v * 256] =
                sqrtf(sxz[v] * sxz[v] + syz[v] * syz[v] + EPSc);
        }
        // publish sigma halos: sxz row 15 (M=15 -> VGPR7, hi half), syz col 15 (n==15 lanes)
        if (hi) haloSxz[tile * 16 + n] = sxz[7];
        if (n == 15) {
#pragma unroll
            for (int v = 0; v < 8; ++v) haloSyz[tile * 16 + mb + v] = syz[v];
        }
        // stage sigma tiles in LDS for phase B operand reads
#pragma unroll
        for (int v = 0; v < 8; ++v) {
            ld0[(mb + v) * 16 + n] = sxz[v];
            ld1[(mb + v) * 16 + n] = syz[v];
        }
        grid_sync(barCnt, barGen);

        // ================= phase B: velocity =================
        asm volatile("s_wait_dscnt 0" ::: "memory");
        v2f bsx[4], asy[4];
#pragma unroll
        for (int k = 0; k < 4; ++k) {
            int r0 = 4 * k + 2 * hi;
            bsx[k].x = ld0[r0 * 16 + n];
            bsx[k].y = ld0[(r0 + 1) * 16 + n];
            asy[k].x = ld1[n * 16 + r0];
            asy[k].y = ld1[n * 16 + r0 + 1];
        }
        v8f dsx = {}, dsy = {};
#pragma unroll
        for (int k = 0; k < 4; ++k)
            dsx = __builtin_amdgcn_wmma_f32_16x16x4_f32(false, aLb[k], false, bsx[k],
                                                        (short)0, dsx, false, false);
#pragma unroll
        for (int k = 0; k < 4; ++k)
            dsy = __builtin_amdgcn_wmma_f32_16x16x4_f32(false, asy[k], false, bRb[k],
                                                        (short)0, dsy, false, false);
        // halo fixes: row 0 gets -sxz(above tile, row15); col 0 gets -syz(left tile, col15)
        if (ti > 0 && !hi) dsx[0] -= haloSxz[((ti - 1) * 16 + tj) * 16 + n];
        if (tj > 0 && n == 0) {
            const float* h = &haloSyz[(ti * 16 + tj - 1) * 16 + mb];
#pragma unroll
            for (int v = 0; v < 8; ++v) dsy[v] -= h[v];
        }

#pragma unroll
        for (int v = 0; v < 8; ++v) {
            float ddx = gxv[v] - xt;
            float ddy = gy - yt;
            float rsq = ddx * ddx + ddy * ddy;
            float src = expf(-rsq / TWOSIG2c);
            float acc = dsx[v] * INV_Dc + dsy[v] * INV_Dc + src;
            float nv  = (vz[v] + DT_RHOc * acc) * BULKFc * damp[v];
            vz[v] = nv;
            vzHist[(size_t)s * 65536 + gbase + v * 256] = nv;
        }
        // publish vz halos for next step: row 0 (M=0 -> VGPR0, lo half), col 0 (n==0 lanes)
        if (!hi) haloVzRow[tile * 16 + n] = vz[0];
        if (n == 0) {
#pragma unroll
            for (int v = 0; v < 8; ++v) haloVzCol[tile * 16 + mb + v] = vz[v];
        }
        grid_sync(barCnt, barGen);
    }
}

extern "C" void kernel_launch(void* const* d_in, const int* in_sizes, int n_in,
                              void* d_out, int out_size, void* d_ws, size_t ws_size,
                              hipStream_t stream) {
    (void)in_sizes; (void)n_in; (void)out_size; (void)ws_size;
    const float* traj = (const float*)d_in[0];   // [300][2] float32
    float* out = (float*)d_out;                  // vz_history ++ stress_history
    float* ws  = (float*)d_ws;                   // halos + barrier state (~64KB)
    wave_init_kernel<<<65, 256, 0, stream>>>(ws);
    wave_persistent_kernel<<<NBLOCKS, 256, 0, stream>>>(traj, out, ws);
}